// DeltaGradientDescent_73916387164402
// MI455X (gfx1250) — compile-verified
//
#include <hip/hip_runtime.h>
#include <math.h>

#define DIM    4096
#define BATCH  64
#define LR     0.001f
#define ALPHA  0.01f
#define EPSV   1e-8f

#define BWAVES 8                 // waves per block
#define KCHUNK (DIM / BWAVES)    // 512: K-range per wave in Phase B

typedef __attribute__((ext_vector_type(2))) float v2f;
typedef __attribute__((ext_vector_type(8))) float v8f;

// ---------------------------------------------------------------------------
// Phase A: row-wise L2 normalization of input_x (BATCH x DIM) -> xn
// One block (256 threads = 8 waves) per row.
// ---------------------------------------------------------------------------
__global__ __launch_bounds__(256)
void k_normalize(const float* __restrict__ x, float* __restrict__ xn) {
    __shared__ float red[8];
    const int row = blockIdx.x;
    const float* xr = x + (size_t)row * DIM;

    float s = 0.f;
    for (int i = threadIdx.x; i < DIM; i += 256) {
        float v = xr[i];
        s += v * v;
    }
    // wave32 reduction
    for (int off = 16; off > 0; off >>= 1)
        s += __shfl_down(s, off, 32);
    const int wave = threadIdx.x >> 5;
    const int lane = threadIdx.x & 31;
    if (lane == 0) red[wave] = s;
    __syncthreads();
    if (wave == 0) {
        float t = (lane < 8) ? red[lane] : 0.f;
        for (int off = 4; off > 0; off >>= 1)
            t += __shfl_down(t, off, 32);
        if (lane == 0) red[0] = 1.0f / (sqrtf(t) + EPSV);
    }
    __syncthreads();
    const float inv = red[0];
    float* xo = xn + (size_t)row * DIM;
    for (int i = threadIdx.x; i < DIM; i += 256)
        xo[i] = xr[i] * inv;
}

// ---------------------------------------------------------------------------
// Phase B: Y = W @ xn^T   (Y: DIM x BATCH), M=4096 N=64 K=4096, f32 WMMA.
// One block per 16-row M-tile (256 blocks -> 256 WGPs active).  The 8 waves
// of a block split K into chunks of 512; each wave accumulates a partial
// 16x64 tile (4 WMMA accumulators), then the block reduces the 8 partials
// through LDS (32 KB) in a fixed order (deterministic).
// ---------------------------------------------------------------------------
__global__ __launch_bounds__(256)
void k_wxT(const float* __restrict__ W, const float* __restrict__ xn,
           float* __restrict__ Y) {
    __shared__ float lds[BWAVES * 16 * BATCH];   // 8 * 1024 f32 = 32 KB

    const int wave  = threadIdx.x >> 5;
    const int lane  = threadIdx.x & 31;
    const int half  = lane >> 4;        // 0: K={0,1}, 1: K={2,3}
    const int l16   = lane & 15;
    const int mBase = blockIdx.x * 16;
    const int kBase = wave * KCHUNK;
    const int kOff  = half * 2;

    // A: W[mBase + l16][kBase + k + kOff .. +1]   (contiguous float2)
    const float* Ap = W + (size_t)(mBase + l16) * DIM + kBase + kOff;
    // B: xn^T -> B[k][col] = xn[col][k]; col = t*16 + l16 (contiguous float2)
    const float* B0 = xn + (size_t)( 0 + l16) * DIM + kBase + kOff;
    const float* B1 = xn + (size_t)(16 + l16) * DIM + kBase + kOff;
    const float* B2 = xn + (size_t)(32 + l16) * DIM + kBase + kOff;
    const float* B3 = xn + (size_t)(48 + l16) * DIM + kBase + kOff;

    v8f acc0 = {}, acc1 = {}, acc2 = {}, acc3 = {};

    #pragma unroll 8
    for (int k = 0; k < KCHUNK; k += 4) {
        v2f a  = *(const v2f*)(Ap + k);
        v2f b0 = *(const v2f*)(B0 + k);
        v2f b1 = *(const v2f*)(B1 + k);
        v2f b2 = *(const v2f*)(B2 + k);
        v2f b3 = *(const v2f*)(B3 + k);
        acc0 = __builtin_amdgcn_wmma_f32_16x16x4_f32(false, a, false, b0,
                                                     (short)0, acc0, false, false);
        acc1 = __builtin_amdgcn_wmma_f32_16x16x4_f32(false, a, false, b1,
                                                     (short)0, acc1, false, false);
        acc2 = __builtin_amdgcn_wmma_f32_16x16x4_f32(false, a, false, b2,
                                                     (short)0, acc2, false, false);
        acc3 = __builtin_amdgcn_wmma_f32_16x16x4_f32(false, a, false, b3,
                                                     (short)0, acc3, false, false);
    }

    // Spill partial tile to LDS.  C/D layout: VGPR r -> (M = r + 8*half, N = l16)
    float* lw = lds + wave * (16 * BATCH);
    #pragma unroll
    for (int r = 0; r < 8; ++r) {
        const int m = r + half * 8;
        float* lp = lw + m * BATCH + l16;
        lp[0]  = acc0[r];
        lp[16] = acc1[r];
        lp[32] = acc2[r];
        lp[48] = acc3[r];
    }
    __syncthreads();

    // Fixed-order cross-wave reduction: 1024 outputs, 4 per thread.
    for (int o = threadIdx.x; o < 16 * BATCH; o += 256) {
        float s = 0.f;
        #pragma unroll
        for (int w = 0; w < BWAVES; ++w)
            s += lds[w * (16 * BATCH) + o];
        const int m = o >> 6;     // / BATCH
        const int b = o & 63;     // % BATCH
        Y[(size_t)(mBase + m) * BATCH + b] = s;
    }
}

// ---------------------------------------------------------------------------
// Phase C: out = W - (alpha/64) * (Y @ xn) - lr * grad
// GEMM M=N=4096, K=64 fused with elementwise update.
// One wave per 16x16 output tile: 65536 waves -> 8192 blocks x 8 waves.
// ---------------------------------------------------------------------------
__global__ __launch_bounds__(256)
void k_update(const float* __restrict__ W, const float* __restrict__ grad,
              const float* __restrict__ Y, const float* __restrict__ xn,
              float* __restrict__ out) {
    const int waveId = blockIdx.x * 8 + (threadIdx.x >> 5);   // 0..65535
    const int lane   = threadIdx.x & 31;
    const int half   = lane >> 4;
    const int l16    = lane & 15;
    const int mTile  = waveId >> 8;       // 0..255
    const int nTile  = waveId & 255;
    const int mBase  = mTile * 16;
    const int nBase  = nTile * 16;

    // A: Y[mBase + l16][k + 2*half .. +1]   (contiguous float2, row stride 64)
    const float* Ap = Y + (size_t)(mBase + l16) * BATCH + half * 2;
    // B: xn[k + 2*half][nBase + l16]        (two strided dword loads)
    const float* Bp = xn + (size_t)(half * 2) * DIM + nBase + l16;

    v8f acc = {};
    #pragma unroll
    for (int k = 0; k < BATCH; k += 4) {
        v2f a = *(const v2f*)(Ap + k);
        v2f b;
        b.x = Bp[(size_t)k * DIM];
        b.y = Bp[(size_t)(k + 1) * DIM];
        acc = __builtin_amdgcn_wmma_f32_16x16x4_f32(false, a, false, b,
                                                    (short)0, acc, false, false);
    }

    const float scale = ALPHA / (float)BATCH;
    #pragma unroll
    for (int r = 0; r < 8; ++r) {
        const int m = mBase + r + half * 8;
        const int n = nBase + l16;
        const size_t idx = (size_t)m * DIM + n;
        out[idx] = W[idx] - scale * acc[r] - LR * grad[idx];
    }
}

// ---------------------------------------------------------------------------
extern "C" void kernel_launch(void* const* d_in, const int* in_sizes, int n_in,
                              void* d_out, int out_size, void* d_ws, size_t ws_size,
                              hipStream_t stream) {
    const float* weight = (const float*)d_in[0];   // (DIM, DIM)
    const float* inputx = (const float*)d_in[1];   // (BATCH, DIM)
    const float* grad   = (const float*)d_in[2];   // (DIM, DIM)
    float* out = (float*)d_out;                    // (DIM, DIM)

    float* xn = (float*)d_ws;                      // BATCH*DIM floats (1 MB)
    float* Y  = xn + (size_t)BATCH * DIM;          // DIM*BATCH floats (1 MB)

    // Phase A: normalize rows of input_x
    k_normalize<<<BATCH, 256, 0, stream>>>(inputx, xn);

    // Phase B: Y = W @ xn^T   (256 blocks x 8 waves, K split across waves)
    k_wxT<<<DIM / 16, 256, 0, stream>>>(weight, xn, Y);

    // Phase C: out = W - (alpha/64) * Y @ xn - lr * grad  (65536 waves)
    k_update<<<(DIM / 16) * (DIM / 16) / 8, 256, 0, stream>>>(weight, grad, Y, xn, out);
}